// CrossAttention_61718680044194
// MI455X (gfx1250) — compile-verified
//
#include <hip/hip_runtime.h>
#include <math.h>

// Problem constants (match reference)
#define B_ 8
#define T_ 128
#define I_ 196
#define TD_ 768
#define ID_ 1024
#define H_ 512

typedef __attribute__((ext_vector_type(2))) float v2f;
typedef __attribute__((ext_vector_type(8))) float v8f;
typedef __attribute__((address_space(3))) float lds_float;

#if __has_builtin(__builtin_amdgcn_tanhf)
#define TANHF(x) __builtin_amdgcn_tanhf(x)
#else
#define TANHF(x) tanhf(x)
#endif

// Async global->LDS copy of one dword per lane (ASYNCcnt-tracked), ISA §15.18.3.
__device__ __forceinline__ void async_copy_b32(const float* gsrc, float* ldst) {
    unsigned lds_addr = (unsigned)(unsigned long)(lds_float*)ldst;
    asm volatile("global_load_async_to_lds_b32 %0, %1, off"
                 :: "v"(lds_addr), "v"(gsrc) : "memory");
}
__device__ __forceinline__ void wait_asynccnt0() {
    asm volatile("s_wait_asynccnt 0x0" ::: "memory");
}

// ---------------------------------------------------------------------------
// Generic batched GEMM: C[bz] = A[bz] (MxK, row major) * Bm[bz] (KxN, row major)
// (+ bias[n] if bias != nullptr).  One wave (32 threads) computes a 16x64
// output strip using 4x V_WMMA_F32_16X16X4_F32, reusing the A fragment.
// Requires K % 4 == 0, N % 64 == 0.  Ragged M handled by clamping the A row
// (garbage rows of D are never stored) so the hot loop has no exec masking.
// ---------------------------------------------------------------------------
__global__ __launch_bounds__(32) void wmma_gemm_f32(
    const float* __restrict__ A, const float* __restrict__ Bm,
    const float* __restrict__ bias, float* __restrict__ C,
    int M, int N, int K, long sA, long sB, long sC)
{
    const int lane = threadIdx.x;      // 0..31
    const int half = lane >> 4;        // 0: K rows {0,1}; 1: K rows {2,3}
    const int l16  = lane & 15;
    const int mTile = blockIdx.y;
    const int nBase = blockIdx.x * 64;
    const long bz = blockIdx.z;

    const float* Ab = A  + bz * sA;
    const float* Bb = Bm + bz * sB;
    float*       Cb = C  + bz * sC;

    const int rowA  = mTile * 16 + l16;          // A-matrix row for this lane
    const int rowLd = (rowA < M) ? rowA : (M - 1);  // clamp: no branch in loop

    // Hoisted, incremented pointers: loop body is pure loads + WMMA.
    const float* ap = Ab + (long)rowLd * K + half * 2;
    const float* bp = Bb + (long)(half * 2) * N + nBase + l16;
    const long bStep = 4L * (long)N;

    v8f acc[4];
#pragma unroll
    for (int j = 0; j < 4; ++j) acc[j] = v8f{};

    for (int k0 = 0; k0 < K; k0 += 4) {
        // A fragment: 16x4, lane half selects K pair -> 8B aligned b64 load
        const v2f afrag = *(const v2f*)ap;
        // B fragments: 4x16 each; lanes stripe N, VGPR index stripes K row
        v2f b0, b1, b2, b3;
        b0.x = bp[0];   b0.y = bp[N];
        b1.x = bp[16];  b1.y = bp[N + 16];
        b2.x = bp[32];  b2.y = bp[N + 32];
        b3.x = bp[48];  b3.y = bp[N + 48];
        __builtin_prefetch(bp + 2 * bStep, 0, 1);   // global_prefetch_b8

        acc[0] = __builtin_amdgcn_wmma_f32_16x16x4_f32(false, afrag, false, b0,
                                                       (short)0, acc[0], false, false);
        acc[1] = __builtin_amdgcn_wmma_f32_16x16x4_f32(false, afrag, false, b1,
                                                       (short)0, acc[1], false, false);
        acc[2] = __builtin_amdgcn_wmma_f32_16x16x4_f32(false, afrag, false, b2,
                                                       (short)0, acc[2], false, false);
        acc[3] = __builtin_amdgcn_wmma_f32_16x16x4_f32(false, afrag, false, b3,
                                                       (short)0, acc[3], false, false);
        ap += 4;
        bp += bStep;
    }

    // C/D layout: lanes 0-15 => N=lane, M=r; lanes 16-31 => N=lane-16, M=8+r
#pragma unroll
    for (int j = 0; j < 4; ++j) {
        const int col = nBase + j * 16 + l16;
        const float bv = bias ? bias[col] : 0.0f;
#pragma unroll
        for (int r = 0; r < 8; ++r) {
            const int row = mTile * 16 + half * 8 + r;
            if (row < M) Cb[(long)row * N + col] = acc[j][r] + bv;
        }
    }
}

// ---------------------------------------------------------------------------
// scores[b,t,i] = sum_h wa[h] * tanh(pt[b,t,h] + pi[b,i,h]) + ba
// One block per (b,t); pt row + wa staged into LDS with async-to-LDS loads
// (ASYNCcnt); one wave per i, lanes split H with v_tanh_f32.
// ---------------------------------------------------------------------------
__global__ __launch_bounds__(256) void scores_kernel(
    const float* __restrict__ pt, const float* __restrict__ pi,
    const float* __restrict__ wa, const float* __restrict__ ba,
    float* __restrict__ scores)
{
    __shared__ float s_pt[H_];
    __shared__ float s_wa[H_];
    const int b = blockIdx.y;
    const int t = blockIdx.x;
    const int tid = threadIdx.x;

    const float* ptrow = pt + ((long)b * T_ + t) * H_;
    async_copy_b32(ptrow + tid,       &s_pt[tid]);
    async_copy_b32(ptrow + tid + 256, &s_pt[tid + 256]);
    async_copy_b32(wa + tid,          &s_wa[tid]);
    async_copy_b32(wa + tid + 256,    &s_wa[tid + 256]);
    wait_asynccnt0();
    __syncthreads();

    const int lane = tid & 31;
    const int wave = tid >> 5;   // 8 waves
    const float bav = ba[0];

    for (int i = wave; i < I_; i += 8) {
        const float* prow = pi + ((long)b * I_ + i) * H_;
        float acc = 0.0f;
#pragma unroll 4
        for (int h = lane; h < H_; h += 32)
            acc += s_wa[h] * TANHF(s_pt[h] + prow[h]);
#pragma unroll
        for (int off = 16; off > 0; off >>= 1)
            acc += __shfl_down(acc, off, 32);
        if (lane == 0)
            scores[((long)b * T_ + t) * I_ + i] = acc + bav;
    }
}

// ---------------------------------------------------------------------------
// Softmax over i (last axis, length 196). One block per (b,t) row.
// ---------------------------------------------------------------------------
__global__ __launch_bounds__(256) void softmax_rows_kernel(
    const float* __restrict__ S, float* __restrict__ P)
{
    __shared__ float red[256];
    const long row = blockIdx.x;
    const int tid = threadIdx.x;
    const float* s = S + row * I_;

    float v = (tid < I_) ? s[tid] : -3.4e38f;
    red[tid] = v; __syncthreads();
#pragma unroll
    for (int o = 128; o > 0; o >>= 1) {
        if (tid < o) red[tid] = fmaxf(red[tid], red[tid + o]);
        __syncthreads();
    }
    const float mx = red[0]; __syncthreads();
    const float e = (tid < I_) ? __expf(v - mx) : 0.0f;
    red[tid] = e; __syncthreads();
#pragma unroll
    for (int o = 128; o > 0; o >>= 1) {
        if (tid < o) red[tid] += red[tid + o];
        __syncthreads();
    }
    const float inv = 1.0f / red[0];
    if (tid < I_) P[row * I_ + tid] = e * inv;
}

// ---------------------------------------------------------------------------
// Softmax over t (axis 1, length 128), written transposed: P2t[b,i,t].
// One block per (b,i), 128 threads (one per t).
// ---------------------------------------------------------------------------
__global__ __launch_bounds__(128) void softmax_cols_kernel(
    const float* __restrict__ S, float* __restrict__ Pt)
{
    __shared__ float red[128];
    const int b = blockIdx.y;
    const int i = blockIdx.x;
    const int t = threadIdx.x;

    const float v = S[((long)b * T_ + t) * I_ + i];
    red[t] = v; __syncthreads();
#pragma unroll
    for (int o = 64; o > 0; o >>= 1) {
        if (t < o) red[t] = fmaxf(red[t], red[t + o]);
        __syncthreads();
    }
    const float mx = red[0]; __syncthreads();
    const float e = __expf(v - mx);
    red[t] = e; __syncthreads();
#pragma unroll
    for (int o = 64; o > 0; o >>= 1) {
        if (t < o) red[t] += red[t + o];
        __syncthreads();
    }
    Pt[((long)b * I_ + i) * T_ + t] = e / red[0];
}

// ---------------------------------------------------------------------------
extern "C" void kernel_launch(void* const* d_in, const int* in_sizes, int n_in,
                              void* d_out, int out_size, void* d_ws, size_t ws_size,
                              hipStream_t stream) {
    (void)in_sizes; (void)n_in; (void)out_size; (void)ws_size;

    const float* text  = (const float*)d_in[0];  // [B,T,768]
    const float* image = (const float*)d_in[1];  // [B,I,1024]
    const float* Wt    = (const float*)d_in[2];  // [768,512]
    const float* bt    = (const float*)d_in[3];  // [512]
    const float* Wi    = (const float*)d_in[4];  // [1024,512]
    const float* bi    = (const float*)d_in[5];  // [512]
    const float* wa    = (const float*)d_in[6];  // [512]
    const float* ba    = (const float*)d_in[7];  // [1]

    // Workspace layout (floats): pt | pi | scores | P1 | P2t  (~7.7 MB)
    float* pt  = (float*)d_ws;
    float* pi  = pt  + (size_t)B_ * T_ * H_;   //  524,288
    float* sc  = pi  + (size_t)B_ * I_ * H_;   //  802,816
    float* P1  = sc  + (size_t)B_ * T_ * I_;   //  200,704
    float* P2t = P1  + (size_t)B_ * T_ * I_;   //  200,704

    float* out_text = (float*)d_out;                      // [B,T,1024]
    float* out_img  = out_text + (size_t)B_ * T_ * ID_;   // [B,I,768]

    // 1) pt = text @ Wt + bt   (M=B*T=1024, N=512, K=768; batch collapsed)
    wmma_gemm_f32<<<dim3(512 / 64, (B_ * T_) / 16, 1), 32, 0, stream>>>(
        text, Wt, bt, pt, B_ * T_, H_, TD_, 0, 0, 0);

    // 2) pi = image @ Wi + bi  (M=B*I=1568, N=512, K=1024)
    wmma_gemm_f32<<<dim3(512 / 64, (B_ * I_) / 16, 1), 32, 0, stream>>>(
        image, Wi, bi, pi, B_ * I_, H_, ID_, 0, 0, 0);

    // 3) scores = wa . tanh(pt + pi) + ba
    scores_kernel<<<dim3(T_, B_), 256, 0, stream>>>(pt, pi, wa, ba, sc);

    // 4) softmax over i -> P1 ; softmax over t (transposed) -> P2t
    softmax_rows_kernel<<<B_ * T_, 256, 0, stream>>>(sc, P1);
    softmax_cols_kernel<<<dim3(I_, B_), 128, 0, stream>>>(sc, P2t);

    // 5) text_attended = P1 @ image   (per b: M=128, N=1024, K=196)
    wmma_gemm_f32<<<dim3(ID_ / 64, T_ / 16, B_), 32, 0, stream>>>(
        P1, image, nullptr, out_text,
        T_, ID_, I_, (long)T_ * I_, (long)I_ * ID_, (long)T_ * ID_);

    // 6) image_attended = P2t @ text  (per b: M=196 ragged, N=768, K=128)
    wmma_gemm_f32<<<dim3(TD_ / 64, (I_ + 15) / 16, B_), 32, 0, stream>>>(
        P2t, text, nullptr, out_img,
        I_, TD_, T_, (long)I_ * T_, (long)T_ * TD_, (long)I_ * TD_);
}